// SoftReordering_8830452760994
// MI455X (gfx1250) — compile-verified
//
#include <hip/hip_runtime.h>
#include <math.h>

typedef __attribute__((ext_vector_type(2))) float v2f;
typedef __attribute__((ext_vector_type(8))) float v8f;
typedef int v4i_vs __attribute__((vector_size(4 * sizeof(int))));

#define BSZ   64
#define SEQ   128
#define EMB   512
#define WINSZ 7
#define FDIM  (WINSZ * EMB)   // 3584
#define KC    256             // K-chunk per LDS stage
#define NCHUNK (FDIM / KC)    // 14
#define SA_STRIDE 260         // padded LDS row stride (floats): 16B-aligned rows,
                              // 260 % 64 == 4 -> fragment lanes hit distinct banks

#if __has_builtin(__builtin_amdgcn_global_load_async_to_lds_b128)
#define HAVE_ASYNC_LDS 1
#else
#define HAVE_ASYNC_LDS 0
#endif

// async builtin wants: (v4i __device__ *src, v4i __local__ *dst, imm off, imm cpol)
#define AS1V4(p) ((__attribute__((address_space(1))) v4i_vs*)(p))
#define AS3V4(p) ((__attribute__((address_space(3))) v4i_vs*)(p))

__device__ __forceinline__ void wait_asynccnt0() {
#if __has_builtin(__builtin_amdgcn_s_wait_asynccnt)
    __builtin_amdgcn_s_wait_asynccnt(0);
#else
    asm volatile("s_wait_asynccnt 0" ::: "memory");
#endif
}

__global__ __launch_bounds__(256)
void soft_reorder_wmma_kernel(const float* __restrict__ x,
                              const float* __restrict__ W,
                              const float* __restrict__ bias,
                              float* __restrict__ out)
{
    // grid: (SEQ, BSZ/16); block: 256 threads = 8 waves (wave32)
    const int s     = blockIdx.x;
    const int mtile = blockIdx.y;          // 16 batch rows per block
    const int t     = threadIdx.x;
    const int wv    = t >> 5;              // wave id 0..7
    const int ln    = t & 31;              // lane in wave
    const int half  = ln >> 4;             // lane half selects K pair
    const int mr    = ln & 15;             // A: M row / B,C: N col

    __shared__ __align__(16) float sA[16 * SA_STRIDE]; // 16 rows x 256 K (padded)
    __shared__ __align__(16) float sW[KC * WINSZ];     // 256 K x 7 N (contiguous)
    __shared__ float sC[8 * 256];                      // per-wave partial C tiles
    __shared__ float sGate[16 * 8];                    // gates, 16 rows x 7 (pad 8)

    v8f c = {};                            // f32 16x16 accumulator (8 VGPRs)

    const size_t wbase = (size_t)s * FDIM * (size_t)WINSZ;

    for (int chunk = 0; chunk < NCHUNK; ++chunk) {
        // ---- stage A chunk: 16 rows x 256 floats ----
        // a 256-K chunk lies inside one window slot: wslot = chunk>>1, so the
        // source is a contiguous 256-float slice of x[row][s-3+wslot].
        // sp is uniform across the block -> uniform branch.
        const int wslot = chunk >> 1;
        const int sp    = s - (WINSZ / 2) + wslot;
        {
            const int m  = t >> 4;
            const int kb = (t & 15) * 16;              // 16 floats per thread
            const int row = mtile * 16 + m;
            const int eb  = (chunk & 1) * 256 + kb;    // embed offset
            float* dst = &sA[m * SA_STRIDE + kb];
            if (sp >= 0 && sp < SEQ) {
                const float* src = x + ((size_t)row * SEQ + sp) * EMB + eb;
#if HAVE_ASYNC_LDS
                #pragma unroll
                for (int i = 0; i < 4; ++i)
                    __builtin_amdgcn_global_load_async_to_lds_b128(
                        AS1V4(src + 4 * i), AS3V4(dst + 4 * i), 0, 0);
#else
                #pragma unroll
                for (int i = 0; i < 4; ++i) {
                    float4 v = ((const float4*)src)[i];
                    dst[4*i + 0] = v.x; dst[4*i + 1] = v.y;
                    dst[4*i + 2] = v.z; dst[4*i + 3] = v.w;
                }
#endif
            } else {
                #pragma unroll
                for (int i = 0; i < 16; ++i) dst[i] = 0.0f;
            }
        }
        // ---- stage W chunk: 256 K x 7 floats, contiguous layout-preserving copy
        {
            const float* wsrc = W + wbase + (size_t)chunk * KC * WINSZ;
#if HAVE_ASYNC_LDS
            // KC*7*4 bytes = 448 b128 transfers, 256 threads
            #pragma unroll
            for (int r = 0; r < 2; ++r) {
                const int idx = t + r * 256;
                if (idx < (KC * WINSZ) / 4)
                    __builtin_amdgcn_global_load_async_to_lds_b128(
                        AS1V4(wsrc + idx * 4), AS3V4(&sW[idx * 4]), 0, 0);
            }
#else
            #pragma unroll
            for (int j = 0; j < WINSZ; ++j)
                sW[t * WINSZ + j] = wsrc[(size_t)t * WINSZ + j];
#endif
        }
        if (t == 0 && chunk + 1 < NCHUNK)
            __builtin_prefetch(&W[wbase + (size_t)(chunk + 1) * KC * WINSZ], 0, 1);
#if HAVE_ASYNC_LDS
        wait_asynccnt0();
#endif
        __syncthreads();

        // ---- WMMA: each wave owns 32 K of this chunk (8 steps of K=4) ----
        const int kwave = wv * 32;
        #pragma unroll
        for (int step = 0; step < 8; ++step) {
            const int klo = kwave + step * 4 + half * 2;   // lane's K pair
            v2f a;
            a.x = sA[mr * SA_STRIDE + klo];
            a.y = sA[mr * SA_STRIDE + klo + 1];
            v2f bf;
            if (mr < WINSZ) {
                bf.x = sW[klo * WINSZ + mr];
                bf.y = sW[(klo + 1) * WINSZ + mr];
            } else {
                bf.x = 0.0f; bf.y = 0.0f;
            }
            // D = A(16x4 f32) * B(4x16 f32) + C(16x16 f32)
            c = __builtin_amdgcn_wmma_f32_16x16x4_f32(
                    false, a, false, bf, (short)0, c, false, false);
        }
        __syncthreads();  // before next chunk overwrites sA/sW
    }

    // ---- reduce 8 per-wave partial C tiles ----
    #pragma unroll
    for (int v = 0; v < 8; ++v)
        sC[wv * 256 + (v + half * 8) * 16 + mr] = c[v];
    __syncthreads();

    {
        float sum = 0.0f;
        #pragma unroll
        for (int w2 = 0; w2 < 8; ++w2) sum += sC[w2 * 256 + t];
        const int m = t >> 4;
        const int n = t & 15;
        if (n < WINSZ) {
            const float g = sum + bias[s * WINSZ + n];
            sGate[m * 8 + n] = 1.0f / (1.0f + __expf(-g));
        }
    }
    __syncthreads();

    // ---- gating pass: out[row][s][e] = tanh(sum_w gate[w] * x[row][s-3+w][e])
    // 16 rows x 512 emb = 8192 elements, coalesced over threads
    #pragma unroll
    for (int i = 0; i < 32; ++i) {
        const int idx = t + i * 256;
        const int m   = idx >> 9;          // row within tile
        const int e   = idx & 511;
        const int row = mtile * 16 + m;
        float acc = 0.0f;
        #pragma unroll
        for (int w = 0; w < WINSZ; ++w) {
            const int spw = s - (WINSZ / 2) + w;
            if (spw >= 0 && spw < SEQ)
                acc += sGate[m * 8 + w] * x[((size_t)row * SEQ + spw) * EMB + e];
        }
        out[((size_t)row * SEQ + s) * EMB + e] = tanhf(acc);
    }
}

extern "C" void kernel_launch(void* const* d_in, const int* in_sizes, int n_in,
                              void* d_out, int out_size, void* d_ws, size_t ws_size,
                              hipStream_t stream) {
    (void)in_sizes; (void)n_in; (void)out_size; (void)d_ws; (void)ws_size;
    const float* x    = (const float*)d_in[0];   // (64,128,512)
    const float* W    = (const float*)d_in[1];   // (128,3584,7)
    const float* bias = (const float*)d_in[2];   // (128,7)
    float* out        = (float*)d_out;           // (64,128,512)

    dim3 grid(SEQ, BSZ / 16);   // (128, 4) -> 512 workgroups
    dim3 block(256);            // 8 wave32 per workgroup
    soft_reorder_wmma_kernel<<<grid, block, 0, stream>>>(x, W, bias, out);
}